// LGNN_layer_20444044329726
// MI455X (gfx1250) — compile-verified
//
#include <hip/hip_runtime.h>
#include <hip/hip_bf16.h>

typedef float v2f __attribute__((ext_vector_type(2)));
typedef float v8f __attribute__((ext_vector_type(8)));

__device__ __forceinline__ void atomAddF(float* p, float v) {
    unsafeAtomicAdd(p, v);   // -> global_atomic_add_f32
}

// ---------------------------------------------------------------------------
// zero (float4 stores), n multiple of 4
// ---------------------------------------------------------------------------
__global__ void zero_f32(float* __restrict__ p, size_t n) {
    float4* p4 = (float4*)p;
    size_t n4 = n >> 2;
    for (size_t i = blockIdx.x * (size_t)blockDim.x + threadIdx.x; i < n4;
         i += (size_t)gridDim.x * blockDim.x) {
        p4[i] = make_float4(0.f, 0.f, 0.f, 0.f);
    }
}

// ---------------------------------------------------------------------------
// WMMA GEMM: out[R,128] = A[R,K] @ Wcat[K,128]
//   Wcat[k][j] = (j<64) ? W1[(wroff+k)*64 + j] : W2[(wroff+k)*64 + j-64]
// Block: 128 threads = 4 waves; each wave computes a 16x128 strip via
// v_wmma_f32_16x16x4_f32 (fp32 in/out, K-step 4).
// LDS weight layout is K-pair interleaved: element (k2, n) = {W[2k2][n],
// W[2k2+1][n]} so each B fragment is one aligned ds_load_b64 straight into
// a WMMA-ready even-aligned VGPR pair (no repack moves).
// R is a multiple of 16 here (50000, 100000), K in {64,128}.
// ---------------------------------------------------------------------------
__global__ __launch_bounds__(128) void gemm_wmma_128(
    const float* __restrict__ A, int K, int R,
    const float* __restrict__ W1, const float* __restrict__ W2, int wroff,
    float* __restrict__ out)
{
    __shared__ __align__(16) float Ws[128 * 128];   // up to 64 KB; K*128 used

    const int tid = threadIdx.x;
    for (int idx = tid; idx < K * 128; idx += 128) {
        const int k = idx >> 7;
        const int j = idx & 127;
        const float w = (j < 64) ? W1[(size_t)(wroff + k) * 64 + j]
                                 : W2[(size_t)(wroff + k) * 64 + (j - 64)];
        // interleaved pair layout: (k/2, j) -> offset (k/2)*256 + j*2 + (k&1)
        Ws[(k >> 1) * 256 + j * 2 + (k & 1)] = w;
    }
    __syncthreads();

    const int wave = tid >> 5;
    const int lane = tid & 31;
    const int rowbase = blockIdx.x * 64 + wave * 16;
    if (rowbase >= R) return;              // wave-uniform: EXEC stays all-ones

    const int half = lane >> 4;            // 0: lanes 0-15, 1: lanes 16-31
    const int l16  = lane & 15;

    v8f c[8];
#pragma unroll
    for (int j = 0; j < 8; ++j)
#pragma unroll
        for (int r = 0; r < 8; ++r) c[j][r] = 0.0f;

    // A fragment base: row = rowbase + l16, k offset = half*2 (K-pairs/half)
    const float* Aptr = A + (size_t)(rowbase + l16) * K + half * 2;
    const v2f* Wp = (const v2f*)Ws;        // v2f element (k2, n) at k2*128 + n
    const int nsteps = K >> 2;

    for (int s = 0; s < nsteps; ++s) {
        v2f a = *(const v2f*)(Aptr + 4 * s);        // A[M][4s+half*2 .. +1]
        const int k2 = 2 * s + half;                // pair index (4s+half*2)/2
#pragma unroll
        for (int j = 0; j < 8; ++j) {
            v2f b = Wp[k2 * 128 + j * 16 + l16];    // {B[kb][N], B[kb+1][N]}
            c[j] = __builtin_amdgcn_wmma_f32_16x16x4_f32(
                false, a, false, b, (short)0, c[j], false, false);
        }
    }

    float* orow = out + (size_t)rowbase * 128;
#pragma unroll
    for (int j = 0; j < 8; ++j)
#pragma unroll
        for (int r = 0; r < 8; ++r)
            orow[(size_t)(r + half * 8) * 128 + j * 16 + l16] = c[j][r];
}

// ---------------------------------------------------------------------------
// Edge-parallel SpMM-accumulate: acc[rows[e], :] += vals[e] * B[cols[e], :]
// One wave per edge; each lane handles 4 consecutive cols (float4 gather,
// 4 native fp32 atomics). Gather source and accumulator stay L2-resident.
// ---------------------------------------------------------------------------
__global__ __launch_bounds__(256) void spmm_acc(
    const int* __restrict__ rows, const int* __restrict__ cols,
    const float* __restrict__ vals, const float* __restrict__ Bm,
    float* __restrict__ acc, int nnz)
{
    const size_t gtid = blockIdx.x * (size_t)blockDim.x + threadIdx.x;
    const int e = (int)(gtid >> 5);
    if (e >= nnz) return;
    const int lane = threadIdx.x & 31;

    const int   r = rows[e];
    const int   c = cols[e];
    const float v = vals[e];

    const float4* src = (const float4*)(Bm + (size_t)c * 128);
    float4 g = src[lane];

    float* dst = acc + (size_t)r * 128 + (size_t)lane * 4;
    atomAddF(dst + 0, v * g.x);
    atomAddF(dst + 1, v * g.y);
    atomAddF(dst + 2, v * g.z);
    atomAddF(dst + 3, v * g.w);
}

// ---------------------------------------------------------------------------
// In-place ReLU on cols [0,64) + accumulate per-column sum / sumsq into
// stats[0:128] / stats[128:256] (f32 atomics; stats pre-zeroed).
// blockDim = 256 -> 2 row-threads per column.
// ---------------------------------------------------------------------------
__global__ __launch_bounds__(256) void relu_stats(
    float* __restrict__ x, int R, float* __restrict__ stats)
{
    __shared__ float sh[256];
    __shared__ float sh2[256];
    const int col = threadIdx.x & 127;
    const int sub = threadIdx.x >> 7;

    float s = 0.f, s2 = 0.f;
    for (int row = blockIdx.x * 2 + sub; row < R; row += gridDim.x * 2) {
        const size_t idx = (size_t)row * 128 + col;
        float v = x[idx];
        if (col < 64) { v = fmaxf(v, 0.f); x[idx] = v; }
        s += v; s2 += v * v;
    }
    sh[threadIdx.x] = s; sh2[threadIdx.x] = s2;
    __syncthreads();
    if (sub == 0) {
        s  = sh[threadIdx.x]  + sh[threadIdx.x + 128];
        s2 = sh2[threadIdx.x] + sh2[threadIdx.x + 128];
        atomAddF(&stats[col], s);
        atomAddF(&stats[128 + col], s2);
    }
}

// ---------------------------------------------------------------------------
// In-place BatchNorm apply: x = g*(x-mu)*rsqrt(var+eps)+b, biased var.
// ---------------------------------------------------------------------------
__global__ __launch_bounds__(256) void bn_apply(
    float* __restrict__ x, int R, const float* __restrict__ stats,
    const float* __restrict__ g, const float* __restrict__ b)
{
    const size_t total = (size_t)R * 128;
    const float invR = 1.0f / (float)R;
    for (size_t idx = blockIdx.x * (size_t)blockDim.x + threadIdx.x; idx < total;
         idx += (size_t)gridDim.x * blockDim.x) {
        const int col = (int)(idx & 127);
        const float mu  = stats[col] * invR;
        float var = stats[128 + col] * invR - mu * mu;
        var = fmaxf(var, 0.0f);
        x[idx] = g[col] * (x[idx] - mu) * rsqrtf(var + 1e-5f) + b[col];
    }
}

// ---------------------------------------------------------------------------
extern "C" void kernel_launch(void* const* d_in, const int* in_sizes, int n_in,
                              void* d_out, int out_size, void* d_ws, size_t ws_size,
                              hipStream_t stream)
{
    const float* node_feat = (const float*)d_in[0];
    const float* line_feat = (const float*)d_in[1];
    const int*   Fa_rows   = (const int*)d_in[2];
    const int*   Fa_cols   = (const int*)d_in[3];
    const float* Fa_vals   = (const float*)d_in[4];
    const int*   Fb_rows   = (const int*)d_in[5];
    const int*   Fb_cols   = (const int*)d_in[6];
    const float* Fb_vals   = (const float*)d_in[7];
    const int*   Pm_rows   = (const int*)d_in[8];
    const int*   Pm_cols   = (const int*)d_in[9];
    const float* Pm_vals   = (const float*)d_in[10];
    const int*   Pd_rows   = (const int*)d_in[11];
    const int*   Pd_cols   = (const int*)d_in[12];
    const float* Pd_vals   = (const float*)d_in[13];
    const float* W_x2x_1   = (const float*)d_in[14];
    const float* W_x2x_2   = (const float*)d_in[15];
    const float* W_y2x_1   = (const float*)d_in[16];
    const float* W_y2x_2   = (const float*)d_in[17];
    const float* W_y2y_1   = (const float*)d_in[18];
    const float* W_y2y_2   = (const float*)d_in[19];
    const float* W_x2y_1   = (const float*)d_in[20];
    const float* W_x2y_2   = (const float*)d_in[21];
    const float* bn_x_g    = (const float*)d_in[22];
    const float* bn_x_b    = (const float*)d_in[23];
    const float* bn_y_g    = (const float*)d_in[24];
    const float* bn_y_b    = (const float*)d_in[25];

    const int N    = in_sizes[0] / 64;
    const int M    = in_sizes[1] / 64;
    const int nnzA = in_sizes[2] / 4;
    const int nnzB = in_sizes[5] / 4;
    const int nnzP = in_sizes[8];

    float* Xpre  = (float*)d_out;                      // [N,128] -> x
    float* Ypre  = (float*)d_out + (size_t)N * 128;    // [M,128] -> y
    float* B     = (float*)d_ws;                       // [M,128] scratch
    float* stats = (float*)d_ws + (size_t)M * 128;     // 256 floats

    auto zero = [&](float* p, size_t n) {
        unsigned blocks = (unsigned)((n / 4 + 255) / 256);
        if (blocks > 4096) blocks = 4096;
        zero_f32<<<dim3(blocks), dim3(256), 0, stream>>>(p, n);
    };
    auto gemm = [&](const float* Ain, int K, int R, const float* W1,
                    const float* W2, int wroff, float* Outp) {
        gemm_wmma_128<<<dim3((unsigned)((R + 63) / 64)), dim3(128), 0, stream>>>(
            Ain, K, R, W1, W2, wroff, Outp);
    };
    auto spmm = [&](const int* r, const int* c, const float* v,
                    const float* Bm, float* acc, int nnz) {
        unsigned blocks = (unsigned)(((size_t)nnz * 32 + 255) / 256);
        spmm_acc<<<dim3(blocks), dim3(256), 0, stream>>>(r, c, v, Bm, acc, nnz);
    };

    // ---------------- Phase 1: node update -> x = BN([relu(z) | z_p]) -------
    zero(Xpre, (size_t)N * 128);
    for (int i = 0; i < 4; ++i) {
        gemm(node_feat, 64, N, W_x2x_1, W_x2x_2, i * 64, B);     // H_i = f @ [W1_i|W2_i]
        spmm(Fa_rows + (size_t)i * nnzA, Fa_cols + (size_t)i * nnzA,
             Fa_vals + (size_t)i * nnzA, B, Xpre, nnzA);
    }
    gemm(line_feat, 64, M, W_y2x_1, W_y2x_2, 0, B);              // G_m
    spmm(Pm_rows, Pm_cols, Pm_vals, B, Xpre, nnzP);
    gemm(line_feat, 64, M, W_y2x_1, W_y2x_2, 64, B);             // G_d
    spmm(Pd_rows, Pd_cols, Pd_vals, B, Xpre, nnzP);

    zero(stats, 256);
    relu_stats<<<dim3(256), dim3(256), 0, stream>>>(Xpre, N, stats);
    bn_apply<<<dim3(2048), dim3(256), 0, stream>>>(Xpre, N, stats, bn_x_g, bn_x_b);

    // ---------------- Phase 2: line-graph update -> y -----------------------
    zero(Ypre, (size_t)M * 128);
    for (int i = 0; i < 4; ++i) {
        gemm(line_feat, 64, M, W_y2y_1, W_y2y_2, i * 64, B);     // K_i
        spmm(Fb_rows + (size_t)i * nnzB, Fb_cols + (size_t)i * nnzB,
             Fb_vals + (size_t)i * nnzB, B, Ypre, nnzB);
    }
    gemm(Xpre, 128, N, W_x2y_1, W_x2y_2, 0, B);                  // Xm = x @ [.|.]
    spmm(Pm_cols, Pm_rows, Pm_vals, B, Ypre, nnzP);              // Pm^T
    gemm(Xpre, 128, N, W_x2y_1, W_x2y_2, 128, B);                // Xd
    spmm(Pd_cols, Pd_rows, Pd_vals, B, Ypre, nnzP);              // Pd^T

    zero(stats, 256);
    relu_stats<<<dim3(256), dim3(256), 0, stream>>>(Ypre, M, stats);
    bn_apply<<<dim3(2048), dim3(256), 0, stream>>>(Ypre, M, stats, bn_y_g, bn_y_b);
}